// MultiBoxLoss_72232759984312
// MI455X (gfx1250) — compile-verified
//
#include <hip/hip_runtime.h>
#include <hip/hip_bf16.h>
#include <stdint.h>

// Problem constants (match reference setup_inputs)
constexpr int BATCH  = 128;
constexpr int NPRIOR = 8732;
constexpr int NTRUTH = 16;
constexpr int NCLS   = 21;
constexpr int NEGPOS = 3;
constexpr float THRESH = 0.5f;

constexpr int NCHUNK = 8;                 // ce_kernel chunks per batch
constexpr int CHUNK  = (NPRIOR + NCHUNK - 1) / NCHUNK;  // 1092
constexpr int TILE   = 256;               // priors per async tile
constexpr int SORT_N = 16384;             // power of two >= NPRIOR

// ---------------- CDNA5 async global -> LDS copy ----------------
#if defined(__has_builtin)
#if __has_builtin(__builtin_amdgcn_global_load_async_to_lds_b32)
#define HAVE_ASYNC_B32 1
#endif
#endif

typedef __attribute__((address_space(1))) int gint_t;
typedef __attribute__((address_space(3))) int lint_t;

__device__ __forceinline__ void async_dword_to_lds(const float* gsrc, float* ldst) {
    // low 32 bits of a generic pointer to __shared__ is the LDS byte offset
    unsigned lds_off = (unsigned)(uintptr_t)ldst;
#if defined(HAVE_ASYNC_B32)
    __builtin_amdgcn_global_load_async_to_lds_b32(
        (gint_t*)(uintptr_t)gsrc, (lint_t*)(uintptr_t)lds_off, 0, 0);
#else
    asm volatile("global_load_async_to_lds_b32 %0, %1, off"
                 :: "v"(lds_off), "v"(gsrc) : "memory");
#endif
}

__device__ __forceinline__ void wait_async_zero() {
    asm volatile("s_wait_asynccnt 0x0" ::: "memory");
}

__device__ __forceinline__ float sl1f(float d) {
    float ad = fabsf(d);
    return ad < 1.f ? 0.5f * ad * ad : ad - 0.5f;
}

// ================= Kernel 1: matching + localization loss =================
__global__ __launch_bounds__(256)
void match_kernel(const float* __restrict__ loc_data,
                  const float* __restrict__ priors,
                  const float* __restrict__ targets,
                  int*   __restrict__ conf_t,
                  float* __restrict__ loc_loss,
                  int*   __restrict__ pos_cnt) {
    __shared__ float         s_bov[NPRIOR];
    __shared__ unsigned char s_bidx[NPRIOR];
    __shared__ float s_tx1[NTRUTH], s_ty1[NTRUTH], s_tx2[NTRUTH], s_ty2[NTRUTH];
    __shared__ float s_lab[NTRUTH], s_area[NTRUTH];
    __shared__ float s_rv[256];
    __shared__ int   s_ri[256];
    __shared__ int   s_bp[NTRUTH];
    __shared__ float s_red[256];
    __shared__ int   s_redi[256];

    const int b = blockIdx.x, tid = threadIdx.x;

    if (tid < NTRUTH) {
        const float* tg = targets + ((size_t)b * NTRUTH + tid) * 5;
        float x1 = tg[0], y1 = tg[1], x2 = tg[2], y2 = tg[3];
        s_tx1[tid] = x1; s_ty1[tid] = y1; s_tx2[tid] = x2; s_ty2[tid] = y2;
        s_lab[tid] = tg[4];
        s_area[tid] = (x2 - x1) * (y2 - y1);
    }
    __syncthreads();

    // Pass 1: per-prior best truth, per-truth best prior (thread-local)
    float tbv[NTRUTH]; int tbi[NTRUTH];
#pragma unroll
    for (int t = 0; t < NTRUTH; ++t) { tbv[t] = -1.f; tbi[t] = 0; }

    for (int p = tid; p < NPRIOR; p += 256) {
        float4 pr = ((const float4*)priors)[p];
        float px1 = pr.x - 0.5f * pr.z, py1 = pr.y - 0.5f * pr.w;
        float px2 = pr.x + 0.5f * pr.z, py2 = pr.y + 0.5f * pr.w;
        float parea = (px2 - px1) * (py2 - py1);
        float best = -1.f; int bi = 0;
#pragma unroll
        for (int t = 0; t < NTRUTH; ++t) {
            float lx = fmaxf(s_tx1[t], px1), ly = fmaxf(s_ty1[t], py1);
            float rx = fminf(s_tx2[t], px2), ry = fminf(s_ty2[t], py2);
            float w = fmaxf(rx - lx, 0.f), h = fmaxf(ry - ly, 0.f);
            float inter = w * h;
            float iou = inter / (s_area[t] + parea - inter);
            if (iou > best) { best = iou; bi = t; }          // first-index tie-break
            if (iou > tbv[t]) { tbv[t] = iou; tbi[t] = p; }  // first-index tie-break
        }
        s_bov[p] = best;
        s_bidx[p] = (unsigned char)bi;
    }
    __syncthreads();

    // Per-truth argmax over priors: block reduce with lowest-index tie-break
    for (int t = 0; t < NTRUTH; ++t) {
        s_rv[tid] = tbv[t]; s_ri[tid] = tbi[t];
        __syncthreads();
        for (int s = 128; s > 0; s >>= 1) {
            if (tid < s) {
                float v2 = s_rv[tid + s]; int i2 = s_ri[tid + s];
                if (v2 > s_rv[tid] || (v2 == s_rv[tid] && i2 < s_ri[tid])) {
                    s_rv[tid] = v2; s_ri[tid] = i2;
                }
            }
            __syncthreads();
        }
        if (tid == 0) s_bp[t] = s_ri[0];
        __syncthreads();
    }

    // Sequential override: later truths win on collision
    if (tid == 0) {
        for (int t = 0; t < NTRUTH; ++t) {
            int bp = s_bp[t];
            s_bov[bp] = 2.f;
            s_bidx[bp] = (unsigned char)t;
        }
    }
    __syncthreads();

    // Pass 2: conf_t, loc encoding, smooth-L1 over positives
    float lloc = 0.f; int cnt = 0;
    for (int p = tid; p < NPRIOR; p += 256) {
        int idx = s_bidx[p];
        float ov = s_bov[p];
        int conf = (ov < THRESH) ? 0 : ((int)s_lab[idx] + 1);
        conf_t[(size_t)b * NPRIOR + p] = conf;
        if (conf > 0) {
            cnt++;
            float4 pr = ((const float4*)priors)[p];
            float mx1 = s_tx1[idx], my1 = s_ty1[idx], mx2 = s_tx2[idx], my2 = s_ty2[idx];
            float gx = ((mx1 + mx2) * 0.5f - pr.x) / (0.1f * pr.z);
            float gy = ((my1 + my2) * 0.5f - pr.y) / (0.1f * pr.w);
            float gw = logf((mx2 - mx1) / pr.z) / 0.2f;
            float gh = logf((my2 - my1) / pr.w) / 0.2f;
            float4 ld = ((const float4*)loc_data)[(size_t)b * NPRIOR + p];
            lloc += sl1f(ld.x - gx) + sl1f(ld.y - gy) + sl1f(ld.z - gw) + sl1f(ld.w - gh);
        }
    }
    s_red[tid] = lloc; s_redi[tid] = cnt;
    __syncthreads();
    for (int s = 128; s > 0; s >>= 1) {
        if (tid < s) { s_red[tid] += s_red[tid + s]; s_redi[tid] += s_redi[tid + s]; }
        __syncthreads();
    }
    if (tid == 0) { loc_loss[b] = s_red[0]; pos_cnt[b] = s_redi[0]; }
}

// ================= Kernel 2: cross entropy via async LDS streaming =================
__global__ __launch_bounds__(256)
void ce_kernel(const float* __restrict__ conf_data,
               const int*   __restrict__ conf_t,
               float* __restrict__ ce_neg,
               float* __restrict__ pos_part) {
    __shared__ float buf[2][TILE * NCLS];
    __shared__ float s_red[256];

    const int b = blockIdx.x, ch = blockIdx.y, tid = threadIdx.x;
    const int c0 = ch * CHUNK;
    const int c1 = (c0 + CHUNK < NPRIOR) ? (c0 + CHUNK) : NPRIOR;
    const int ntiles = (c1 - c0 + TILE - 1) / TILE;

    // issue tile 0
    {
        int t0 = c0;
        int cnt = (c1 - t0 < TILE) ? (c1 - t0) : TILE;
        const float* g = conf_data + ((size_t)b * NPRIOR + t0) * NCLS;
        for (int i = tid; i < cnt * NCLS; i += 256)
            async_dword_to_lds(g + i, &buf[0][i]);
    }

    float lpos = 0.f;
    for (int tile = 0; tile < ntiles; ++tile) {
        wait_async_zero();      // this wave's in-flight tile has landed
        __syncthreads();        // all waves' data landed; prev buffer free

        if (tile + 1 < ntiles) {
            int t0 = c0 + (tile + 1) * TILE;
            int cnt = (c1 - t0 < TILE) ? (c1 - t0) : TILE;
            const float* g = conf_data + ((size_t)b * NPRIOR + t0) * NCLS;
            float* dst = buf[(tile + 1) & 1];
            for (int i = tid; i < cnt * NCLS; i += 256)
                async_dword_to_lds(g + i, dst + i);
        }

        const int cur = tile & 1;
        const int t0 = c0 + tile * TILE;
        const int cnt = (c1 - t0 < TILE) ? (c1 - t0) : TILE;
        if (tid < cnt) {
            const int p = t0 + tid;
            const float* x = &buf[cur][tid * NCLS];
            float m = x[0];
#pragma unroll
            for (int c = 1; c < NCLS; ++c) m = fmaxf(m, x[c]);
            float s = 0.f;
#pragma unroll
            for (int c = 0; c < NCLS; ++c) s += expf(x[c] - m);
            int ct = conf_t[(size_t)b * NPRIOR + p];
            float ce = (m + logf(s)) - x[ct];
            if (ct > 0) {
                lpos += ce;
                ce_neg[(size_t)b * NPRIOR + p] = 0.f;
            } else {
                ce_neg[(size_t)b * NPRIOR + p] = ce;
            }
        }
        __syncthreads();
    }

    s_red[tid] = lpos;
    __syncthreads();
    for (int s = 128; s > 0; s >>= 1) {
        if (tid < s) s_red[tid] += s_red[tid + s];
        __syncthreads();
    }
    if (tid == 0) pos_part[b * NCHUNK + ch] = s_red[0];
}

// ================= Kernel 3: per-batch top-k sum (bitonic sort in LDS) =================
__global__ __launch_bounds__(1024)
void topk_kernel(const float* __restrict__ ce_neg,
                 const int*   __restrict__ pos_cnt,
                 float* __restrict__ neg_loss) {
    __shared__ float s[SORT_N];  // exactly 64 KB; partials reuse s[0..1023]
    const int b = blockIdx.x, tid = threadIdx.x;

    for (int i = tid; i < SORT_N; i += 1024)
        s[i] = (i < NPRIOR) ? ce_neg[(size_t)b * NPRIOR + i] : -1e30f;
    __syncthreads();

    // ascending bitonic sort; top-k = tail
    for (unsigned k = 2; k <= SORT_N; k <<= 1) {
        for (unsigned j = k >> 1; j > 0; j >>= 1) {
            for (unsigned i = tid; i < SORT_N; i += 1024) {
                unsigned ixj = i ^ j;
                if (ixj > i) {
                    float a = s[i], c = s[ixj];
                    bool up = ((i & k) == 0);
                    if (up ? (a > c) : (a < c)) { s[i] = c; s[ixj] = a; }
                }
            }
            __syncthreads();
        }
    }

    int k = NEGPOS * pos_cnt[b];
    if (k > NPRIOR - 1) k = NPRIOR - 1;
    float acc = 0.f;
    // reads are at indices >= SORT_N-1-(NPRIOR-2) = 7653 > 1023, so safe to
    // reuse s[0..1023] for partials after a barrier
    for (int i = tid; i < k; i += 1024) acc += s[SORT_N - 1 - i];
    __syncthreads();
    s[tid] = acc;
    __syncthreads();
    for (int st = 512; st > 0; st >>= 1) {
        if (tid < st) s[tid] += s[tid + st];
        __syncthreads();
    }
    if (tid == 0) neg_loss[b] = s[0];
}

// ================= Kernel 4: WMMA-based final reduction =================
typedef __attribute__((ext_vector_type(2))) float v2f;
typedef __attribute__((ext_vector_type(8))) float v8f;

// Sum one float per lane across the full wave32 using V_WMMA_F32_16X16X4_F32.
// A = all ones (16x4), B carries the 32 values (other 32 B-slots = 0), C = 0.
// D[i][j] = sum_k B[k][j]; summing D row 0 over its 16 columns = sum of all
// B entries = sum of the 32 lane values (layout-agnostic).
__device__ __forceinline__ float wmma_sum32(float v) {
    v2f a; a.x = 1.f; a.y = 1.f;
    v2f bb; bb.x = v; bb.y = 0.f;
    v8f c = {0.f, 0.f, 0.f, 0.f, 0.f, 0.f, 0.f, 0.f};
    c = __builtin_amdgcn_wmma_f32_16x16x4_f32(false, a, false, bb,
                                              (short)0, c, false, false);
    float x = c[0];  // lanes 0..15: col sums j=0..15 (lanes 16..31: same sums, row 8)
    x += __shfl_xor(x, 8, 32);
    x += __shfl_xor(x, 4, 32);
    x += __shfl_xor(x, 2, 32);
    x += __shfl_xor(x, 1, 32);
    return x;  // lanes 0..15 hold the total
}

__global__ __launch_bounds__(32)
void finalize_kernel(const float* __restrict__ loc_loss,
                     const int*   __restrict__ pos_cnt,
                     const float* __restrict__ pos_part,
                     const float* __restrict__ neg_loss,
                     float* __restrict__ out) {
    const int lane = threadIdx.x;  // single full wave32: EXEC all ones for WMMA
    float l1 = 0.f, n = 0.f, lp = 0.f, ln = 0.f;
    for (int i = lane; i < BATCH; i += 32) {  // uniform trip count: no divergence
        l1 += loc_loss[i];
        n  += (float)pos_cnt[i];
        ln += neg_loss[i];
    }
    for (int i = lane; i < BATCH * NCHUNK; i += 32) lp += pos_part[i];

    float L1 = wmma_sum32(l1);
    float N  = wmma_sum32(n);
    float LP = wmma_sum32(lp);
    float LN = wmma_sum32(ln);
    if (lane == 0) {
        out[0] = L1 / N;
        out[1] = (LP + LN) / N;
    }
}

// ================= Host launcher =================
extern "C" void kernel_launch(void* const* d_in, const int* in_sizes, int n_in,
                              void* d_out, int out_size, void* d_ws, size_t ws_size,
                              hipStream_t stream) {
    const float* loc     = (const float*)d_in[0];  // (128, 8732, 4)
    const float* conf    = (const float*)d_in[1];  // (128, 8732, 21)
    const float* priors  = (const float*)d_in[2];  // (8732, 4)
    const float* targets = (const float*)d_in[3];  // (128, 16, 5)
    float* out = (float*)d_out;                    // 2 scalars

    constexpr size_t CONF_BYTES = (size_t)BATCH * NPRIOR * 4;
    char* ws = (char*)d_ws;
    int*   conf_t   = (int*)  (ws);
    float* ce_neg   = (float*)(ws + CONF_BYTES);
    float* loc_loss = (float*)(ws + 2 * CONF_BYTES);
    int*   pos_cnt  = (int*)  (ws + 2 * CONF_BYTES + 512);
    float* pos_part = (float*)(ws + 2 * CONF_BYTES + 1024);
    float* neg_loss = (float*)(ws + 2 * CONF_BYTES + 1024 + (size_t)BATCH * NCHUNK * 4);

    match_kernel<<<dim3(BATCH), dim3(256), 0, stream>>>(loc, priors, targets,
                                                        conf_t, loc_loss, pos_cnt);
    ce_kernel<<<dim3(BATCH, NCHUNK), dim3(256), 0, stream>>>(conf, conf_t,
                                                             ce_neg, pos_part);
    topk_kernel<<<dim3(BATCH), dim3(1024), 0, stream>>>(ce_neg, pos_cnt, neg_loss);
    finalize_kernel<<<dim3(1), dim3(32), 0, stream>>>(loc_loss, pos_cnt, pos_part,
                                                      neg_loss, out);
}